// QSA_26147760898095
// MI455X (gfx1250) — compile-verified
//
#include <hip/hip_runtime.h>
#include <math.h>

#define NQ   10
#define DIM  1024
#define HID  64
#define BT   2048   // B*T = 4*512

typedef __attribute__((ext_vector_type(2))) float v2f;
typedef __attribute__((ext_vector_type(8))) float v8f;
typedef __attribute__((ext_vector_type(4))) int   v4i;

#define QSA_GAS __attribute__((address_space(1)))
#define QSA_LAS __attribute__((address_space(3)))

#if defined(__has_builtin)
#  if __has_builtin(__builtin_amdgcn_global_load_async_to_lds_b128)
#    define QSA_ASYNC_LDS 1
#  endif
#endif
#ifndef QSA_ASYNC_LDS
#  define QSA_ASYNC_LDS 0
#endif

// ---------------------------------------------------------------- U = I
__global__ void qsa_init_identity(float* __restrict__ Ur, float* __restrict__ Ui) {
    int idx = blockIdx.x * blockDim.x + threadIdx.x;      // DIM*DIM threads
    int r = idx >> 10, c = idx & (DIM - 1);
    Ur[idx] = (r == c) ? 1.f : 0.f;
    Ui[idx] = 0.f;
}

// ------------------------------------------------- gate table (20 x 2x2 cplx)
// gates[j]    (j<10)  : G0 = RY(ry0[j]) @ RX(rx0[j])
// gates[10+j]         : G1 = RY(ry1[j])
// layout per gate: {g00r,g00i, g01r,g01i, g10r,g10i, g11r,g11i}
__global__ void qsa_prep_gates(const float* __restrict__ rx0,
                               const float* __restrict__ ry0,
                               const float* __restrict__ ry1,
                               float* __restrict__ g) {
    int j = threadIdx.x;
    if (j < NQ) {
        float s1, c1, s2, c2;
        sincosf(rx0[j] * 0.5f, &s1, &c1);
        sincosf(ry0[j] * 0.5f, &s2, &c2);
        float* p = g + j * 8;
        p[0] =  c2 * c1; p[1] =  s2 * s1;   // g00 = c2c1 + i s2s1
        p[2] = -s2 * c1; p[3] = -c2 * s1;   // g01 = -s2c1 - i c2s1
        p[4] =  s2 * c1; p[5] = -c2 * s1;   // g10 =  s2c1 - i c2s1
        p[6] =  c2 * c1; p[7] = -s2 * s1;   // g11 = c2c1 - i s2s1
    } else if (j < 2 * NQ) {
        float s2, c2;
        sincosf(ry1[j - NQ] * 0.5f, &s2, &c2);
        float* p = g + j * 8;
        p[0] = c2;  p[1] = 0.f; p[2] = -s2; p[3] = 0.f;
        p[4] = s2;  p[5] = 0.f; p[6] =  c2; p[7] = 0.f;
    }
}

// ---------------------------------------- single-qubit gate: U <- (I⊗g⊗I) U
// row bit for qubit q is (NQ-1-q). One thread per (row-pair, column).
__global__ void qsa_apply_su2(float* __restrict__ Ur, float* __restrict__ Ui,
                              const float* __restrict__ g8, int bit) {
    int idx = blockIdx.x * blockDim.x + threadIdx.x;      // (DIM/2)*DIM threads
    int c = idx & (DIM - 1);
    int p = idx >> 10;                                     // 0..511
    int low = p & (bit - 1);
    int r0  = ((p - low) << 1) | low;                      // insert 0 at `bit`
    int r1  = r0 | bit;
    float g00r = g8[0], g00i = g8[1], g01r = g8[2], g01i = g8[3];
    float g10r = g8[4], g10i = g8[5], g11r = g8[6], g11i = g8[7];
    int i0 = r0 * DIM + c, i1 = r1 * DIM + c;
    float ar = Ur[i0], ai = Ui[i0], br = Ur[i1], bi = Ui[i1];
    float n0r = g00r*ar - g00i*ai + g01r*br - g01i*bi;
    float n0i = g00r*ai + g00i*ar + g01r*bi + g01i*br;
    float n1r = g10r*ar - g10i*ai + g11r*br - g11i*bi;
    float n1i = g10r*ai + g10i*ar + g11r*bi + g11i*br;
    Ur[i0] = n0r; Ui[i0] = n0i;
    Ur[i1] = n1r; Ui[i1] = n1i;
}

// ---------------------------------------- CNOT: swap rows (c=1,t=0)<->(c=1,t=1)
__global__ void qsa_apply_cnot(float* __restrict__ Ur, float* __restrict__ Ui,
                               int cbit, int tbit) {
    int idx = blockIdx.x * blockDim.x + threadIdx.x;      // (DIM/4)*DIM threads
    int c = idx & (DIM - 1);
    int p = idx >> 10;                                     // 0..255
    int blo = (cbit < tbit) ? cbit : tbit;
    int bhi = cbit ^ tbit ^ blo;
    int l0 = p & (blo - 1);
    int t  = ((p - l0) << 1) | l0;                         // insert 0 at blo
    int l1 = t & (bhi - 1);
    int r  = ((t - l1) << 1) | l1;                         // insert 0 at bhi
    int i0 = (r | cbit) * DIM + c;                         // tbit == 0 here
    int i1 = i0 + tbit * DIM;
    float a = Ur[i0], b = Ui[i0];
    Ur[i0] = Ur[i1]; Ui[i0] = Ui[i1];
    Ur[i1] = a;      Ui[i1] = b;
}

// ---------------------------------------------------------- psi = row-norm(x)
__global__ void qsa_normalize(const float* __restrict__ x, float* __restrict__ psi) {
    __shared__ float red[8];
    int bt = blockIdx.x;
    const float* row = x + (size_t)bt * DIM;
    float s = 0.f;
    for (int i = threadIdx.x; i < DIM; i += 256) { float v = row[i]; s += v * v; }
    for (int off = 16; off > 0; off >>= 1) s += __shfl_xor(s, off, 32);
    int wave = threadIdx.x >> 5, lane = threadIdx.x & 31;
    if (lane == 0) red[wave] = s;
    __syncthreads();
    if (threadIdx.x == 0) {
        float t = 0.f;
        for (int i = 0; i < 8; ++i) t += red[i];
        red[0] = 1.0f / sqrtf(t);
    }
    __syncthreads();
    float inv = red[0];
    for (int i = threadIdx.x; i < DIM; i += 256) psi[(size_t)bt * DIM + i] = row[i] * inv;
}

// --------------------------------------------- phi = psi @ U^T  (fp32 WMMA)
// phiR[b,d] = sum_k psi[b,k]*Ur[d,k] ; phiI likewise with Ui.
// Block = 8 waves sharing one 64-wide N-group; U tile chunk (64 cols x 64 k,
// re+im) staged in LDS once per K-chunk -> 8x B reuse.  Wave tile: M=16 (b),
// N=4x16=64 (d), both re & im -> A fragment reused 8x.
#define KC      64
#define LDSTR   68            // row stride in floats: 16B aligned, bank-safe
__global__ void qsa_gemm_phi(const float* __restrict__ psi,
                             const float* __restrict__ Ur,
                             const float* __restrict__ Ui,
                             float* __restrict__ phiR,
                             float* __restrict__ phiI) {
    __shared__ float BsR[64 * LDSTR];
    __shared__ float BsI[64 * LDSTR];

    const int tid  = threadIdx.x;
    const int mb   = blockIdx.x >> 4;      // 0..15 : 128-row M block
    const int ng   = blockIdx.x & 15;      // 0..15 : 64-col  N group
    const int wave = tid >> 5;             // 0..7
    const int lane = tid & 31;
    const int lm   = lane & 15;
    const int lh   = lane >> 4;

    // staging assignment: thread -> (row of 64, 16-float column slice)
    const int srow = tid >> 2;             // 0..63
    const int scol = (tid & 3) * 16;       // 0,16,32,48

    const v8f zero = {0.f,0.f,0.f,0.f,0.f,0.f,0.f,0.f};
    v8f accR[4] = {zero, zero, zero, zero};
    v8f accI[4] = {zero, zero, zero, zero};

    const float* aRow = psi + (size_t)(mb * 128 + wave * 16 + lm) * DIM;
    float* lR = &BsR[srow * LDSTR + scol];
    float* lI = &BsI[srow * LDSTR + scol];
    const size_t gOff = (size_t)(ng * 64 + srow) * DIM + scol;

    for (int kb = 0; kb < DIM; kb += KC) {
        // ---- stage U chunk (re+im) into LDS
        const float* gR = Ur + gOff + kb;
        const float* gI = Ui + gOff + kb;
#if QSA_ASYNC_LDS
#pragma unroll
        for (int u = 0; u < 4; ++u) {
            __builtin_amdgcn_global_load_async_to_lds_b128(
                (QSA_GAS v4i*)(gR + u * 4), (QSA_LAS v4i*)(lR + u * 4), 0, 0);
            __builtin_amdgcn_global_load_async_to_lds_b128(
                (QSA_GAS v4i*)(gI + u * 4), (QSA_LAS v4i*)(lI + u * 4), 0, 0);
        }
#  if __has_builtin(__builtin_amdgcn_s_wait_asynccnt)
        __builtin_amdgcn_s_wait_asynccnt(0);
#  else
        asm volatile("s_wait_asynccnt 0" ::: "memory");
#  endif
#else
#pragma unroll
        for (int u = 0; u < 4; ++u) {
            float4 vR = *(const float4*)(gR + u * 4);
            float4 vI = *(const float4*)(gI + u * 4);
            *(float4*)(lR + u * 4) = vR;
            *(float4*)(lI + u * 4) = vI;
        }
#endif
        __syncthreads();

        // ---- compute on the staged chunk
        for (int kk = 0; kk < KC; kk += 4) {
            const int col  = kb + kk + 2 * lh;
            const int lcol = kk + 2 * lh;
            // A frag (16x4): lane lm -> M, half-wave lh picks K pair {col,col+1}
            v2f a = *(const v2f*)(aRow + col);
#pragma unroll
            for (int t = 0; t < 4; ++t) {
                const int n = t * 16 + lm;          // B frag (4x16): lane lm -> N
                v2f br = *(const v2f*)(&BsR[n * LDSTR + lcol]);
                v2f bi = *(const v2f*)(&BsI[n * LDSTR + lcol]);
                accR[t] = __builtin_amdgcn_wmma_f32_16x16x4_f32(
                    false, a, false, br, (short)0, accR[t], false, false);
                accI[t] = __builtin_amdgcn_wmma_f32_16x16x4_f32(
                    false, a, false, bi, (short)0, accI[t], false, false);
            }
        }
        __syncthreads();
    }

    // C/D layout: VGPR i -> M = i + 8*lh, N = lm  (within 16x16 tile)
#pragma unroll
    for (int t = 0; t < 4; ++t) {
        const int n = ng * 64 + t * 16 + lm;
#pragma unroll
        for (int i = 0; i < 8; ++i) {
            const size_t row = (size_t)(mb * 128 + wave * 16 + i + 8 * lh);
            phiR[row * DIM + n] = accR[t][i];
            phiI[row * DIM + n] = accI[t][i];
        }
    }
}

// ------------------------- exps[b,h] = Re sum_d conj(phi_d) c_h(d) phi_{d^xm}
// Pauli string O_h: k = d ^ xmask; phase = (-i)^nY * (-1)^popc(d & zymask).
__global__ void qsa_obs(const float* __restrict__ phiR,
                        const float* __restrict__ phiI,
                        const int* __restrict__ op,
                        float* __restrict__ out) {
    __shared__ float sR[DIM];
    __shared__ float sI[DIM];
    const int bt = blockIdx.x;
    for (int i = threadIdx.x; i < DIM; i += 256) {
        sR[i] = phiR[(size_t)bt * DIM + i];
        sI[i] = phiI[(size_t)bt * DIM + i];
    }
    __syncthreads();
    const int wave = threadIdx.x >> 5, lane = threadIdx.x & 31;
    for (int hh = 0; hh < 8; ++hh) {
        const int h = wave * 8 + hh;
        int xm = 0, zym = 0, ny = 0;
        for (int q = 0; q < NQ; ++q) {
            const int code = op[h * NQ + q];
            const int bit = 1 << (NQ - 1 - q);
            if (code == 1 || code == 2) xm |= bit;          // X or Y flips
            if (code == 2) { ++ny; zym |= bit; }            // Y
            if (code == 3) zym |= bit;                      // Z
        }
        float acc = 0.f;
        for (int d = lane; d < DIM; d += 32) {
            const int k = d ^ xm;
            const float pr = sR[d], pi = sI[d], qr = sR[k], qi = sI[k];
            const float s  = (__popc(d & zym) & 1) ? -1.f : 1.f;
            const float tre = pr * qr + pi * qi;            // Re(conj(p)*q)
            const float tim = pr * qi - pi * qr;            // Im(conj(p)*q)
            float v;
            switch (ny & 3) {                                // multiply by (-i)^nY
                case 0:  v = tre;  break;
                case 1:  v = tim;  break;
                case 2:  v = -tre; break;
                default: v = -tim; break;
            }
            acc += s * v;
        }
        for (int off = 16; off > 0; off >>= 1) acc += __shfl_xor(acc, off, 32);
        if (lane == 0) out[(size_t)bt * HID + h] = acc;
    }
}

// ----------------------------------------------------------------- launcher
extern "C" void kernel_launch(void* const* d_in, const int* in_sizes, int n_in,
                              void* d_out, int out_size, void* d_ws, size_t ws_size,
                              hipStream_t stream) {
    const float* x   = (const float*)d_in[0];   // (B,T,DIM)
    const float* rx0 = (const float*)d_in[1];   // (10,)
    const float* ry0 = (const float*)d_in[2];
    const float* ry1 = (const float*)d_in[3];
    const int*   opc = (const int*)d_in[4];     // (64,10)
    float* out = (float*)d_out;                 // (B,T,64)

    float* ws    = (float*)d_ws;
    float* Ur    = ws;                      // 1M floats
    float* Ui    = Ur   + DIM * DIM;        // 1M
    float* psi   = Ui   + DIM * DIM;        // 2M
    float* phiR  = psi  + BT * DIM;         // 2M
    float* phiI  = phiR + BT * DIM;         // 2M
    float* gates = phiI + BT * DIM;         // 160 floats

    // 1) U = I
    qsa_init_identity<<<(DIM * DIM) / 256, 256, 0, stream>>>(Ur, Ui);
    // 2) gate table
    qsa_prep_gates<<<1, 32, 0, stream>>>(rx0, ry0, ry1, gates);
    // 3) layer 1: fused RY(ry0)·RX(rx0) per qubit
    for (int j = 0; j < NQ; ++j)
        qsa_apply_su2<<<(DIM / 2 * DIM) / 256, 256, 0, stream>>>(
            Ur, Ui, gates + j * 8, 1 << (NQ - 1 - j));
    // 4) CNOT ring (control j, target (j+1)%N)
    for (int j = 0; j < NQ; ++j) {
        const int cb = 1 << (NQ - 1 - j);
        const int tb = 1 << (NQ - 1 - ((j + 1) % NQ));
        qsa_apply_cnot<<<(DIM / 4 * DIM) / 256, 256, 0, stream>>>(Ur, Ui, cb, tb);
    }
    // 5) layer 2: RY(ry1)
    for (int j = 0; j < NQ; ++j)
        qsa_apply_su2<<<(DIM / 2 * DIM) / 256, 256, 0, stream>>>(
            Ur, Ui, gates + (NQ + j) * 8, 1 << (NQ - 1 - j));
    // 6) amplitude encoding
    qsa_normalize<<<BT, 256, 0, stream>>>(x, psi);
    // 7) phi = psi @ U^T via fp32 WMMA, LDS-staged U tiles (16 Mblk x 16 Ngrp)
    qsa_gemm_phi<<<256, 256, 0, stream>>>(psi, Ur, Ui, phiR, phiI);
    // 8) Pauli-string expectation values
    qsa_obs<<<BT, 256, 0, stream>>>(phiR, phiI, opc, out);
}